// MyModel_61933428410954
// MI455X (gfx1250) — compile-verified
//
#include <hip/hip_runtime.h>

#define THREADS 256
#define EPT 4        // elements per thread: 4 elems * 3 floats = 48 B = 3 x b128
#define WAVE 32      // gfx1250 is wave32

// Pointee type the builtin expects: 4 x i32 vector (b128 payload)
typedef int v4i __attribute__((vector_size(16)));
typedef __attribute__((address_space(1))) v4i* v4i_global;
typedef __attribute__((address_space(3))) v4i* v4i_lds;

// ---------- CDNA5 async global -> LDS staging (ASYNCcnt path) ----------
__device__ __forceinline__ void async_g2l_b128(const float4* g, float4* l) {
#if __has_builtin(__builtin_amdgcn_global_load_async_to_lds_b128)
  __builtin_amdgcn_global_load_async_to_lds_b128(
      (v4i_global)const_cast<float4*>(g),
      (v4i_lds)l,
      /*offset=*/0, /*cpol=*/0);
#else
  unsigned lds_addr =
      (unsigned)(unsigned long long)((__attribute__((address_space(3))) float4*)l);
  asm volatile("global_load_async_to_lds_b128 %0, %1, off"
               :
               : "v"(lds_addr), "v"(g)
               : "memory");
#endif
}

__device__ __forceinline__ void wait_async0() {
#if __has_builtin(__builtin_amdgcn_s_wait_asynccnt)
  __builtin_amdgcn_s_wait_asynccnt(0);
#else
  asm volatile("s_wait_asynccnt 0" ::: "memory");
#endif
  asm volatile("" ::: "memory");  // compiler-only ordering fence for the LDS reads
}

// ---------- circuit math ----------
__device__ __forceinline__ float lse2(float a, float b) {
  float m = fmaxf(a, b);
  return m + __logf(1.0f + __expf(-fabsf(a - b)));
}

__device__ __forceinline__ float eval_elem(float x0, float x1, float x2) {
  // literals: log(1-p) = log1p(-exp(x)); p in [0.05,0.95] => no cancellation
  float c0 = __logf(1.0f - __expf(x0));
  float c1 = __logf(1.0f - __expf(x1));
  float c2 = __logf(1.0f - __expf(x2));
  float d5 = lse2(x1 + c2, x2 + c1);     // node 5: children (1,2) | (3,4)
  float d7 = x1 + x2;                    // node 7: LSE(x1+x2, -2000) == x1+x2 (f32 exact)
  return lse2(x0 + d5, c0 + d7);         // node 8: children (0,5) | (6,7)
}

__global__ __launch_bounds__(THREADS) void pc_logsumexp_kernel(
    const float* __restrict__ lp, float* __restrict__ out, int n) {
  // 12 KB staging tile, float4-typed for guaranteed 16 B alignment
  __shared__ float4 tile[THREADS * 3];
  const int t = threadIdx.x;
  const int lane = t & (WAVE - 1);
  const int wave = t >> 5;

  // wave-level tiles: 32 lanes * 4 elems = 128 elements = 96 float4s per wave
  const long long waveElemBase =
      (long long)blockIdx.x * (THREADS * EPT) + (long long)wave * (WAVE * EPT);

  if (waveElemBase + WAVE * EPT <= (long long)n) {   // wave-uniform guard
    const long long waveF4 =
        (long long)blockIdx.x * (THREADS * 3) + (long long)wave * (WAVE * 3);
    const float4* g = (const float4*)lp + waveF4;
    float4* lbase = &tile[wave * (WAVE * 3)];

    // perfectly coalesced: each instruction loads 32 lane-contiguous float4s
    async_g2l_b128(g + lane,            lbase + lane);
    async_g2l_b128(g + lane + WAVE,     lbase + lane + WAVE);
    async_g2l_b128(g + lane + 2 * WAVE, lbase + lane + 2 * WAVE);
    wait_async0();  // producers are lanes of this wave -> no workgroup barrier

    const float4* l = lbase + lane * 3;  // this lane's 48 B window
    float4 A = l[0];
    float4 Bv = l[1];
    float4 C = l[2];

    float4 r;
    r.x = eval_elem(A.x, A.y, A.z);
    r.y = eval_elem(A.w, Bv.x, Bv.y);
    r.z = eval_elem(Bv.z, Bv.w, C.x);
    r.w = eval_elem(C.y, C.z, C.w);

    ((float4*)out)[(long long)blockIdx.x * THREADS + t] = r;
  } else {
    // scalar tail (not hit for B = 4,194,304, but keeps the kernel general)
    long long e0 = waveElemBase + (long long)lane * EPT;
    long long e1 = e0 + EPT;
    if (e1 > (long long)n) e1 = (long long)n;
    for (long long e = e0; e < e1; ++e) {
      out[e] = eval_elem(lp[3 * e + 0], lp[3 * e + 1], lp[3 * e + 2]);
    }
  }
}

extern "C" void kernel_launch(void* const* d_in, const int* in_sizes, int n_in,
                              void* d_out, int out_size, void* d_ws, size_t ws_size,
                              hipStream_t stream) {
  (void)n_in; (void)out_size; (void)d_ws; (void)ws_size;
  const float* lp = (const float*)d_in[0];
  float* out = (float*)d_out;
  const int n = in_sizes[0] / 3;          // (B,3) row-major -> B elements
  const int epb = THREADS * EPT;          // 1024 elements per block
  const int blocks = (n + epb - 1) / epb; // 4096 blocks for B = 4,194,304
  pc_logsumexp_kernel<<<blocks, THREADS, 0, stream>>>(lp, out, n);
}